// Decoder_27195732918783
// MI455X (gfx1250) — compile-verified
//
#include <hip/hip_runtime.h>
#include <math.h>

typedef float v2f __attribute__((ext_vector_type(2)));
typedef float v8f __attribute__((ext_vector_type(8)));

#define B_      64
#define U_      4096
#define H_      64
#define T_TOTAL (B_ * U_)   // 262144 recurrence steps / output rows

// ---------------------------------------------------------------------------
// Phase 1: x[t, :] = emb[ids[t]] @ W_ih^T + (b_ih + b_hh), written into d_out.
// ids[t] = y[t % 64, t / 64]  (time-major flatten of y.T)
// One block = 16 output rows; 4 waves = the four 16-wide N tiles.
// fp32 WMMA 16x16x4, K-loop of 16 chunks.  (Verified: emits 16 v_wmma,
// ds_load_2addr_b64 A-fetches pipelined between WMMAs.)
// ---------------------------------------------------------------------------
__global__ __launch_bounds__(128) void rnn_input_proj_wmma(
    const int*   __restrict__ y,
    const float* __restrict__ emb,
    const float* __restrict__ W_ih,
    const float* __restrict__ b_ih,
    const float* __restrict__ b_hh,
    float*       __restrict__ x_out)
{
    // A tile staged in LDS, stride 66 floats to avoid bank conflicts on the
    // column-pair reads (bank = (m*66 + k) % 64 = (2m + k) % 64, distinct per m).
    __shared__ float As[16 * 66];

    const int tid     = threadIdx.x;
    const int rowTile = blockIdx.x;          // 16 rows per block

    // Cooperative gather: 16 embedding rows -> LDS (8 floats per thread).
    {
        const int r  = tid >> 3;             // 0..15
        const int c0 = (tid & 7) * 8;        // 0,8,...,56
        const int t  = rowTile * 16 + r;
        const int id = y[(t & (B_ - 1)) * U_ + (t >> 6)];
        const float* src = emb + (long)id * H_ + c0;
        #pragma unroll
        for (int i = 0; i < 8; ++i) As[r * 66 + c0 + i] = src[i];
    }
    __syncthreads();

    const int wave = tid >> 5;               // 0..3 -> N tile
    const int lane = tid & 31;
    const int m16  = lane & 15;              // M row (A) / N col (B,C)
    const int hi   = lane >> 4;              // 0/1: K half / upper C rows
    const int n    = wave * 16 + m16;        // global output column

    // Preload all 16 B fragments. B[k][n] = W_ih[n, k] (x = e @ W_ih^T).
    // Fragment layout: VGPR v holds row K = 2*hi + v, col = lane%16.
    v2f bfrag[16];
    #pragma unroll
    for (int kc = 0; kc < 16; ++kc) {
        const int k = kc * 4 + 2 * hi;
        bfrag[kc].x = W_ih[n * H_ + k];
        bfrag[kc].y = W_ih[n * H_ + k + 1];
    }

    v8f c = {};
    #pragma unroll
    for (int kc = 0; kc < 16; ++kc) {
        const int k = kc * 4 + 2 * hi;
        v2f a;                                // A: row = m16, K pair per half-wave
        a.x = As[m16 * 66 + k];
        a.y = As[m16 * 66 + k + 1];
        c = __builtin_amdgcn_wmma_f32_16x16x4_f32(
                /*neg_a=*/false, a, /*neg_b=*/false, bfrag[kc],
                /*c_mod=*/(short)0, c, /*reuse_a=*/false, /*reuse_b=*/false);
    }

    const float bias = b_ih[n] + b_hh[n];
    const long  base = (long)rowTile * 16 * H_;
    #pragma unroll
    for (int r = 0; r < 8; ++r) {            // C VGPR r -> row r + 8*hi
        const int m = r + 8 * hi;
        x_out[base + (long)m * H_ + n] = c[r] + bias;
    }
}

// ---------------------------------------------------------------------------
// Fast tanh: gfx1250 has V_TANH_F32 (TRANS32 op). Prefer the clang builtin
// (compiler handles the TRANS co-execution hazard); fall back to inline asm
// with explicit v_nop padding per ISA §7.4.3 (a trans result needs one
// intervening op before its use).
// ---------------------------------------------------------------------------
static __device__ __forceinline__ float fast_tanh(float x) {
#if __has_builtin(__builtin_amdgcn_tanhf)
    return __builtin_amdgcn_tanhf(x);
#elif __has_builtin(__builtin_amdgcn_tanh_f32)
    return __builtin_amdgcn_tanh_f32(x);
#else
    float r;
    asm("v_tanh_f32 %0, %1\n\tv_nop\n\tv_nop" : "=v"(r) : "v"(x));
    return r;
#endif
}

// ---------------------------------------------------------------------------
// Phase 2: strictly serial scan  h = tanh(x_t + W_hh @ h), 262144 steps.
// 64 threads (2 waves): lane j owns output element j, W_hh row j lives in
// 64 VGPRs, h ping-pongs in LDS (broadcast b128 reads), one barrier/step.
// Dot product uses 4 independent accumulators to break the FMA dependency
// chain (issue-limited instead of latency-limited).
// x_t is read from d_out and overwritten in place with h_t.
// ---------------------------------------------------------------------------
__global__ __launch_bounds__(64) void rnn_scan(
    const float* __restrict__ W_hh,
    const float* __restrict__ h0,
    float*       __restrict__ out)
{
    __shared__ __align__(16) float hbuf[2][H_];
    const int j = threadIdx.x;               // 0..63

    // W_hh row j -> registers (new_h[j] = sum_k W_hh[j,k] * h[k]).
    float w[H_];
    #pragma unroll
    for (int k = 0; k < 16; ++k) {
        const float4 t = ((const float4*)(W_hh + j * H_))[k];
        w[4*k+0] = t.x; w[4*k+1] = t.y; w[4*k+2] = t.z; w[4*k+3] = t.w;
    }

    hbuf[0][j] = h0[j];
    __syncthreads();

    float xnext = out[j];                    // prefetch x_0
    for (int t = 0; t < T_TOTAL; ++t) {
        float a0 = xnext, a1 = 0.f, a2 = 0.f, a3 = 0.f;
        if (t + 1 < T_TOTAL)                 // prefetch next x (slot not yet overwritten)
            xnext = out[(long)(t + 1) * H_ + j];

        const float4* hv = (const float4*)hbuf[t & 1];
        #pragma unroll
        for (int k = 0; k < 16; ++k) {       // 16 ds_load_b128 broadcasts
            const float4 h4 = hv[k];
            a0 += w[4*k+0] * h4.x;           // 4 independent FMA chains
            a1 += w[4*k+1] * h4.y;
            a2 += w[4*k+2] * h4.z;
            a3 += w[4*k+3] * h4.w;
        }

        const float hn = fast_tanh((a0 + a1) + (a2 + a3));
        out[(long)t * H_ + j]   = hn;        // overwrite x slot with output
        hbuf[(t & 1) ^ 1][j]    = hn;        // ping-pong: write other buffer
        __syncthreads();                     // single barrier per step
    }
}

extern "C" void kernel_launch(void* const* d_in, const int* in_sizes, int n_in,
                              void* d_out, int out_size, void* d_ws, size_t ws_size,
                              hipStream_t stream) {
    (void)in_sizes; (void)n_in; (void)out_size; (void)d_ws; (void)ws_size;
    const int*   y     = (const int*)  d_in[0];
    const float* emb   = (const float*)d_in[1];
    const float* W_ih  = (const float*)d_in[2];
    const float* W_hh  = (const float*)d_in[3];
    const float* b_ih  = (const float*)d_in[4];
    const float* b_hh  = (const float*)d_in[5];
    const float* h0    = (const float*)d_in[6];
    float* out = (float*)d_out;

    rnn_input_proj_wmma<<<T_TOTAL / 16, 128, 0, stream>>>(y, emb, W_ih, b_ih, b_hh, out);
    rnn_scan<<<1, 64, 0, stream>>>(W_hh, h0, out);
}